// FACActiveContour_44925357916855
// MI455X (gfx1250) — compile-verified
//
#include <hip/hip_runtime.h>
#include <cstdint>

// ---------------------------------------------------------------------------
// TV-L1 style iteration (FAC active contour), 4 x 512 x 512 f32, 100 iters.
// One fused kernel per iteration; state ping-pongs in d_ws (L2-resident).
// p1/p2 stored padded (518x518, 3-px zero border) so TDM tile loads need no
// clamping and zero 'SAME' padding is exact.
// ---------------------------------------------------------------------------

#define IW   512
#define B_   3
#define PP   518            // padded width/height = 512 + 2*3
#define T_   64             // output tile
#define PW   70             // p tile  = T + 6  (halo 3)
#define RW   68             // r tile  = T + 4  (halo 2)
#define NW   66             // np tile = T + 2  (halo 1)
#define THR  0.1f           // LAMBDA * THETA

typedef unsigned int uint32;

// ----------------------------- TDM tile load --------------------------------
#if defined(__HIP_DEVICE_COMPILE__) && defined(__has_builtin)
#if __has_builtin(__builtin_amdgcn_tensor_load_to_lds)
#define HAS_TDM 1
#else
#warning "gfx1250 TDM builtin unavailable: falling back to VGPR-staged tile loads"
#endif
#endif

#ifdef HAS_TDM
typedef unsigned int v4u_t __attribute__((ext_vector_type(4)));
typedef int          v8i_t __attribute__((ext_vector_type(8)));
typedef int          v4i_t __attribute__((ext_vector_type(4)));

// 2-D TDM load: tile_w x tile_h f32 rectangle (row stride = stride elems)
// from global `gaddr` into LDS byte offset `lds_off` (rows stored contiguous,
// pitch = tile_w). tens_w/tens_h = remaining tensor extent (for OOB zeros).
__device__ __forceinline__ void tdm_load_2d(uint32 lds_off, const float* gaddr,
                                            int tile_w, int tile_h,
                                            int tens_w, int tens_h, int stride) {
  unsigned long long ga = (unsigned long long)(uintptr_t)gaddr;
  v4u_t g0;
  g0.x = 1u;                                           // count=1 (valid D#)
  g0.y = lds_off;                                      // lds_addr [63:32]
  g0.z = (uint32)(ga & 0xFFFFFFFFu);                   // global_addr lo
  g0.w = (uint32)((ga >> 32) & 0x1FFFFFFu) | (2u << 30); // addr hi | type=2
  v8i_t g1;
  g1[0] = (2 << 16);                                   // data_size = 4 bytes
  g1[1] = (tens_w & 0xFFFF) << 16;                     // abar=0 | tensor_dim0 lo
  g1[2] = ((tens_w >> 16) & 0xFFFF) | ((tens_h & 0xFFFF) << 16);
  g1[3] = ((tens_h >> 16) & 0xFFFF) | ((tile_w & 0xFFFF) << 16);
  g1[4] = (tile_h & 0xFFFF);                           // tile_dim1 | tile_dim2=0
  g1[5] = stride;                                      // tensor_dim0_stride lo32
  g1[6] = 0;
  g1[7] = 0;
  v4i_t z4 = (v4i_t)0;
#if __clang_major__ >= 23
  v8i_t z8 = (v8i_t)0;
  __builtin_amdgcn_tensor_load_to_lds(g0, g1, z4, z4, z8, 0);
#else
  __builtin_amdgcn_tensor_load_to_lds(g0, g1, z4, z4, 0);
#endif
}
#endif // HAS_TDM

// ------------------------------ init kernels --------------------------------
__global__ void zero_kernel(float* __restrict__ p, size_t n) {
  size_t i = (size_t)blockIdx.x * blockDim.x + threadIdx.x;
  size_t stride = (size_t)gridDim.x * blockDim.x;
  for (; i < n; i += stride) p[i] = 0.0f;
}

__device__ __forceinline__ float ldimg(const float* __restrict__ p, int b, int x, int y) {
  return ((unsigned)x < (unsigned)IW && (unsigned)y < (unsigned)IW)
             ? p[((size_t)b << 18) + ((size_t)y << 9) + x] : 0.0f;
}

// ginv = 1 + gx^2 + gy^2   (g = 1/ginv); iteration-invariant.
__global__ void ginv_kernel(const float* __restrict__ f, float* __restrict__ gv) {
  int idx = blockIdx.x * blockDim.x + threadIdx.x;   // 4*512*512 threads
  int b = idx >> 18;
  int rem = idx & ((1 << 18) - 1);
  int y = rem >> 9, x = rem & 511;
  float a0 = ldimg(f, b, x - 1, y - 1), a2 = ldimg(f, b, x + 1, y - 1);
  float b0 = ldimg(f, b, x - 1, y    ), b2 = ldimg(f, b, x + 1, y    );
  float c0 = ldimg(f, b, x - 1, y + 1), c2 = ldimg(f, b, x + 1, y + 1);
  float a1 = ldimg(f, b, x,     y - 1), c1 = ldimg(f, b, x,     y + 1);
  float gx = (a2 - a0) + 2.0f * (b2 - b0) + (c2 - c0);   // SOBEL_X cross-corr
  float gy = (c0 - a0) + 2.0f * (c1 - a1) + (c2 - a2);   // SOBEL_Y cross-corr
  gv[idx] = 1.0f + gx * gx + gy * gy;
}

// ---------------------------- fused iteration -------------------------------
__global__ __launch_bounds__(256)
void tv_iter_kernel(const float* __restrict__ f,
                    const float* __restrict__ p1s, const float* __restrict__ p2s,
                    const float* __restrict__ vs,  const float* __restrict__ gv,
                    float* __restrict__ p1d, float* __restrict__ p2d,
                    float* __restrict__ vd,  float* __restrict__ uout,
                    int write_u) {
  __shared__ float sp1[PW * PW];
  __shared__ float sp2[PW * PW];
  __shared__ float sr [RW * RW];

  const int tid = threadIdx.x;
  const int ox = blockIdx.x * T_, oy = blockIdx.y * T_, b = blockIdx.z;
  const size_t PADI = (size_t)PP * PP;
  const float* p1img = p1s + (size_t)b * PADI;
  const float* p2img = p2s + (size_t)b * PADI;
  const size_t ibase = (size_t)b << 18;               // b*512*512

  // ---- Phase 1: issue async p1/p2 halo tile DMA (70x70, padded origin) ----
#ifdef HAS_TDM
  {
    // scalar wave id -> scalar branch: exactly one TDM issue per tile
    const int wave = __builtin_amdgcn_readfirstlane((int)(threadIdx.x >> 5));
    const float* g1 = p1img + (size_t)oy * PP + ox;
    const float* g2 = p2img + (size_t)oy * PP + ox;
    if (wave == 0)
      tdm_load_2d((uint32)(uintptr_t)(void*)sp1, g1, PW, PW, PP - ox, PP - oy, PP);
    else if (wave == 1)
      tdm_load_2d((uint32)(uintptr_t)(void*)sp2, g2, PW, PW, PP - ox, PP - oy, PP);
  }
#else
  for (int idx = tid; idx < PW * PW; idx += 256) {
    int i = idx % PW, j = idx / PW;
    size_t g = (size_t)(oy + j) * PP + (ox + i);
    sp1[idx] = p1img[g];
    sp2[idx] = p2img[g];
  }
#endif

  // ---- Phase 2a (overlaps DMA): sr = v - f  over 68x68; prefetch ginv ----
  for (int idx = tid; idx < RW * RW; idx += 256) {
    int i = idx % RW, j = idx / RW;
    int gx = ox - 2 + i, gy = oy - 2 + j;
    float rv = 0.0f;
    if ((unsigned)gx < (unsigned)IW && (unsigned)gy < (unsigned)IW) {
      size_t gi = ibase + ((size_t)gy << 9) + gx;
      rv = vs[gi] - f[gi];                             // -(f - v), THETA = 1
      __builtin_prefetch(&gv[gi], 0, 1);               // global_prefetch for phase 3
    }
    sr[idx] = rv;
  }

#ifdef HAS_TDM
  __builtin_amdgcn_s_wait_tensorcnt(0);                // no-op on non-issuing waves
#endif
  __syncthreads();

  // ---- Phase 2b: sr += div(p)  (same idx->thread map as 2a, no race) ----
  for (int idx = tid; idx < RW * RW; idx += 256) {
    int i = idx % RW, j = idx / RW;
    int gx = ox - 2 + i, gy = oy - 2 + j;
    if ((unsigned)gx < (unsigned)IW && (unsigned)gy < (unsigned)IW) {
      int li = i + 1, lj = j + 1;
      float divp = sp1[lj * PW + li + 1] - sp1[lj * PW + li - 1]
                 + sp2[(lj + 1) * PW + li] - sp2[(lj - 1) * PW + li];
      sr[idx] += divp;                                 // r = div_p - (f - v)
    }
  }
  __syncthreads();

  // ---- Phase 3: p-update (in place in sp1/sp2) over 66x66 (halo 1) ----
  for (int idx = tid; idx < NW * NW; idx += 256) {
    int i = idx % NW, j = idx / NW;
    int gx = ox - 1 + i, gy = oy - 1 + j;
    if ((unsigned)gx < (unsigned)IW && (unsigned)gy < (unsigned)IW) {
      int ri = i + 1, rj = j + 1;
      float t0 = sr[(rj - 1) * RW + ri + 1] - sr[(rj - 1) * RW + ri - 1];
      float t1 = sr[ rj      * RW + ri + 1] - sr[ rj      * RW + ri - 1];
      float t2 = sr[(rj + 1) * RW + ri + 1] - sr[(rj + 1) * RW + ri - 1];
      float rgx = 0.125f * (t0 + 2.0f * t1 + t2);      // SOBEL_X/8
      float s0 = sr[(rj + 1) * RW + ri - 1] - sr[(rj - 1) * RW + ri - 1];
      float s1 = sr[(rj + 1) * RW + ri    ] - sr[(rj - 1) * RW + ri    ];
      float s2 = sr[(rj + 1) * RW + ri + 1] - sr[(rj - 1) * RW + ri + 1];
      float rgy = 0.125f * (s0 + 2.0f * s1 + s2);      // SOBEL_Y/8
      float mag = __builtin_amdgcn_sqrtf(rgx * rgx + rgy * rgy);
      size_t gi = ibase + ((size_t)gy << 9) + gx;
      float denom = 1.0f + 0.125f * mag * gv[gi];
      float inv = __builtin_amdgcn_rcpf(denom);
      int li = i + 2, lj = j + 2;
      sp1[lj * PW + li] = (sp1[lj * PW + li] + 0.125f * rgx) * inv;
      sp2[lj * PW + li] = (sp2[lj * PW + li] + 0.125f * rgy) * inv;
    } // out-of-image: keep padded zero (correct 'SAME' padding)
  }
  __syncthreads();

  // ---- Phase 4: div(p_new), v shrinkage, (u on last iter) over 64x64 ----
  float* p1do = p1d + (size_t)b * PADI;
  float* p2do = p2d + (size_t)b * PADI;
  for (int idx = tid; idx < T_ * T_; idx += 256) {
    int x = idx & (T_ - 1), y = idx >> 6;
    int gx = ox + x, gy = oy + y;
    int li = x + 3, lj = y + 3;
    float divp = sp1[lj * PW + li + 1] - sp1[lj * PW + li - 1]
               + sp2[(lj + 1) * PW + li] - sp2[(lj - 1) * PW + li];
    size_t gi = ibase + ((size_t)gy << 9) + gx;
    float vq = vs[gi];
    float diff = vq + divp;                            // diff = f - u
    float vn = diff > THR ? diff - THR : (diff < -THR ? diff + THR : 0.0f);
    vd[gi] = vn;
    size_t pi = (size_t)(gy + B_) * PP + (gx + B_);
    p1do[pi] = sp1[lj * PW + li];
    p2do[pi] = sp2[lj * PW + li];
    if (write_u) uout[gi] = f[gi] - vq - divp;         // u = f - v_old - div_p
  }
}

// ------------------------------- launcher -----------------------------------
extern "C" void kernel_launch(void* const* d_in, const int* in_sizes, int n_in,
                              void* d_out, int out_size, void* d_ws, size_t ws_size,
                              hipStream_t stream) {
  (void)in_sizes; (void)n_in; (void)out_size; (void)ws_size;
  const float* f = (const float*)d_in[0];
  float* out = (float*)d_out;
  float* ws = (float*)d_ws;

  const size_t PADT = (size_t)PP * PP * 4;   // padded array, 4 images
  const size_t VT   = (size_t)IW * IW * 4;   // plain array, 4 images

  float* p1[2] = { ws,            ws + PADT     };
  float* p2[2] = { ws + 2 * PADT, ws + 3 * PADT };
  float* v [2] = { ws + 4 * PADT, ws + 4 * PADT + VT };
  float* gv    =   ws + 4 * PADT + 2 * VT;

  // zero p1[0..1], p2[0..1] (incl. borders) and v[0] — contiguous region
  zero_kernel<<<2048, 256, 0, stream>>>(ws, 4 * PADT + VT);
  ginv_kernel<<<(int)(VT / 256), 256, 0, stream>>>(f, gv);

  dim3 grid(IW / T_, IW / T_, 4), block(256);
  for (int it = 0; it < 100; ++it) {
    int s = it & 1, d = s ^ 1;
    tv_iter_kernel<<<grid, block, 0, stream>>>(f, p1[s], p2[s], v[s], gv,
                                               p1[d], p2[d], v[d], out,
                                               it == 99 ? 1 : 0);
  }
}